// SSMLayer_59828894433822
// MI455X (gfx1250) — compile-verified
//
#include <hip/hip_runtime.h>
#include <math.h>

// SSM layer for MI455X (gfx1250): fp32 WMMA (16x16x4) GEMMs with fused
// einsum epilogues (D/C/Bm tensors never materialized; ~44 GFLOP fp32,
// ~30MB HBM traffic -> compute bound). Round-2 changes:
//  * cd_kernel: M=64 register blocking -> 4 independent WMMA accumulation
//    chains per wave sharing one B load (4x less w_D fetch per WMMA, and
//    WMMA RAW latency hidden by chain interleaving).
//  * sequential scan replaced by 3-phase chunked affine scan (parallel
//    16x16 chunk composition -> 64-step combine -> parallel replay),
//    cutting the sequential critical path ~20x.

#define Hdim 128
#define Nst  16
#define EPSV 1e-5f
#define LDA  132    // padded LDS row stride
#define MBLK 4      // M-blocking factor in cd_kernel (64 timesteps/block)
#define CLEN 32     // scan chunk length

typedef __attribute__((ext_vector_type(2))) float v2f;
typedef __attribute__((ext_vector_type(8))) float v8f;

// ---------------------------------------------------------------------------
// One 16x16 fp32 output tile, K=128, via 32 chained V_WMMA_F32_16X16X4_F32.
// ---------------------------------------------------------------------------
__device__ __forceinline__ v8f wmma_tile_k128(const float* __restrict__ aLds,
                                              const float* __restrict__ W,
                                              long ldw, long colBase, v8f acc) {
  const int lane = threadIdx.x & 31;
  const int m  = lane & 15;
  const int kh = (lane >> 4) << 1;   // 0 or 2
  const float* wp = W + colBase + (lane & 15);
#pragma unroll
  for (int k = 0; k < Hdim; k += 4) {
    v2f a, b;
    a.x = aLds[m * LDA + k + kh];
    a.y = aLds[m * LDA + k + kh + 1];
    b.x = wp[(long)(k + kh) * ldw];
    b.y = wp[(long)(k + kh + 1) * ldw];
    acc = __builtin_amdgcn_wmma_f32_16x16x4_f32(false, a, false, b,
                                                (short)0, acc, false, false);
  }
  return acc;
}

__device__ __forceinline__ void loadTileRows(const float* __restrict__ A,
                                             int rowBase, int nrows, float* aLds) {
  for (int idx = threadIdx.x; idx < nrows * Hdim; idx += blockDim.x) {
    int r = idx >> 7, c = idx & (Hdim - 1);
    aLds[r * LDA + c] = A[(long)(rowBase + r) * Hdim + c];
  }
}

__device__ __forceinline__ float rowReduce16(float v) {
  v += __shfl_xor(v, 1, 16);
  v += __shfl_xor(v, 2, 16);
  v += __shfl_xor(v, 4, 16);
  v += __shfl_xor(v, 8, 16);
  return v;
}

__device__ __forceinline__ float clip_dt(const float* dtp) {
  return fminf(fmaxf(dtp[0], 1e-4f), 0.1f);
}

// ---------------------------------------------------------------------------
// K1: LayerNorm. One block (128 threads) per row.
// ---------------------------------------------------------------------------
__global__ void ln_kernel(const float* __restrict__ x, const float* __restrict__ g,
                          const float* __restrict__ b, float* __restrict__ xn) {
  __shared__ float red[Hdim];
  const int row = blockIdx.x;
  const int t = threadIdx.x;
  const float v = x[(long)row * Hdim + t];
  red[t] = v;
  __syncthreads();
  for (int s = 64; s > 0; s >>= 1) { if (t < s) red[t] += red[t + s]; __syncthreads(); }
  const float mu = red[0] * (1.0f / Hdim);
  __syncthreads();
  const float d = v - mu;
  red[t] = d * d;
  __syncthreads();
  for (int s = 64; s > 0; s >>= 1) { if (t < s) red[t] += red[t + s]; __syncthreads(); }
  const float var = red[0] * (1.0f / Hdim);
  xn[(long)row * Hdim + t] = d * rsqrtf(var + EPSV) * g[t] + b[t];
}

// ---------------------------------------------------------------------------
// Generic WMMA GEMM out = epi(A @ W + bias), K=128. 8 waves, 1 tile each.
// EPI: 1 gelu(tanh), 2 exp(dt_c * v), 3 + residual.
// ---------------------------------------------------------------------------
template <int EPI>
__global__ void gemm_k128(const float* __restrict__ A, const float* __restrict__ W,
                          const float* __restrict__ bias, const float* __restrict__ resid,
                          float* __restrict__ out, int Ncols, const float* __restrict__ dtp) {
  __shared__ float aLds[16 * LDA];
  const int rowBase = blockIdx.x * 16;
  loadTileRows(A, rowBase, 16, aLds);
  __syncthreads();

  const int wave = threadIdx.x >> 5;
  const int lane = threadIdx.x & 31;
  const long colBase = (long)(blockIdx.y * 8 + wave) * 16;

  v8f acc = {};
  acc = wmma_tile_k128(aLds, W, (long)Ncols, colBase, acc);

  float dtc = 0.0f;
  if (EPI == 2) dtc = clip_dt(dtp);
  const int n = lane & 15;
  const int roff = (lane >> 4) << 3;
#pragma unroll
  for (int r = 0; r < 8; ++r) {
    const long row = rowBase + r + roff;
    const long col = colBase + n;
    float v = acc[r] + bias[col];
    if (EPI == 1) {
      const float c = v + 0.044715f * v * v * v;
      v = 0.5f * v * (1.0f + tanhf(0.7978845608028654f * c));
    } else if (EPI == 2) {
      v = expf(dtc * v);
    } else if (EPI == 3) {
      v += resid[row * Ncols + col];
    }
    out[row * Ncols + col] = v;
  }
}

// ---------------------------------------------------------------------------
// Bu[t,n] = dt_c * sum_h (u@w_B + b_B)[t, n*H+h] * u[t,h]   (Bm fused away)
// ---------------------------------------------------------------------------
__global__ void bu_kernel(const float* __restrict__ u, const float* __restrict__ wB,
                          const float* __restrict__ bB, float* __restrict__ Bu,
                          const float* __restrict__ dtp) {
  __shared__ float uLds[16 * LDA];
  __shared__ float red[8][16];
  const int rowBase = blockIdx.x * 16;
  const int nIdx = blockIdx.y;
  loadTileRows(u, rowBase, 16, uLds);
  __syncthreads();

  const int wave = threadIdx.x >> 5;
  const int lane = threadIdx.x & 31;
  const long colBase = (long)nIdx * Hdim + wave * 16;

  v8f acc = {};
  acc = wmma_tile_k128(uLds, wB, (long)Nst * Hdim, colBase, acc);

  const int j = lane & 15;
  const int roff = (lane >> 4) << 3;
  const float bb = bB[colBase + j];
#pragma unroll
  for (int r = 0; r < 8; ++r) {
    const int rr = r + roff;
    float val = (acc[r] + bb) * uLds[rr * LDA + wave * 16 + j];
    val = rowReduce16(val);
    if (j == 0) red[wave][rr] = val;
  }
  __syncthreads();
  if (threadIdx.x < 16) {
    float s = 0.0f;
#pragma unroll
    for (int w = 0; w < 8; ++w) s += red[w][threadIdx.x];
    Bu[(long)(rowBase + threadIdx.x) * Nst + nIdx] = clip_dt(dtp) * s;
  }
}

// ---------------------------------------------------------------------------
// Chunked affine scan. Phase A: each wave composes one CLEN-step chunk into
// (M_c, v_c): column-per-lane layout, s <- A_d s + Bu  =>  M<-A_d M, v<-A_d v+Bu.
// ---------------------------------------------------------------------------
__global__ void scan_phaseA(const float* __restrict__ Ad, const float* __restrict__ Bu,
                            float* __restrict__ Mc, float* __restrict__ vc, int Lr) {
  __shared__ float aLds[Nst * Nst];
  __shared__ float buLds[Nst];
  const int chunk = blockIdx.x;
  const int ncPerB = Lr / CLEN;
  const int b = chunk / ncPerB;
  const int c = chunk % ncPerB;
  const long t0 = (long)b * Lr + (long)c * CLEN;
  const int lane = threadIdx.x;       // 32 threads
  const int col = lane & 15;
  const bool isV = (lane >= 16);      // lane 16 tracks v; 17..31 shadow it
  float m[Nst];
#pragma unroll
  for (int i = 0; i < Nst; ++i) m[i] = 0.0f;
  if (!isV) m[col] = 1.0f;            // M = I, v = 0
  for (int t = 0; t < CLEN; ++t) {
    const long idx = t0 + t;
    for (int q = lane; q < Nst * Nst; q += 32) aLds[q] = Ad[idx * (Nst * Nst) + q];
    if (lane < Nst) buLds[lane] = Bu[idx * Nst + lane];
    __syncthreads();
    float nm[Nst];
#pragma unroll
    for (int i = 0; i < Nst; ++i) {
      float acc = isV ? buLds[i] : 0.0f;
#pragma unroll
      for (int k = 0; k < Nst; ++k) acc += aLds[i * Nst + k] * m[k];
      nm[i] = acc;
    }
#pragma unroll
    for (int i = 0; i < Nst; ++i) m[i] = nm[i];
    __syncthreads();
  }
  if (!isV) {
#pragma unroll
    for (int i = 0; i < Nst; ++i) Mc[(long)chunk * (Nst * Nst) + i * Nst + col] = m[i];
  } else if (lane == 16) {
#pragma unroll
    for (int i = 0; i < Nst; ++i) vc[(long)chunk * Nst + i] = m[i];
  }
}

// Phase B: sequential combine over chunks (one wave per batch, 64 steps);
// records each chunk's starting state.
__global__ void scan_phaseB(const float* __restrict__ Mc, const float* __restrict__ vc,
                            float* __restrict__ cstart, int ncPerB) {
  const int b = threadIdx.x >> 5;
  const int lane = threadIdx.x & 31;
  const int i = lane & 15;
  float s = 0.0f;
  for (int c = 0; c < ncPerB; ++c) {
    const long ch = (long)b * ncPerB + c;
    if (lane < 16) cstart[ch * Nst + i] = s;
    float acc = vc[ch * Nst + i];
    const float* mrow = Mc + ch * (Nst * Nst) + (long)i * Nst;
#pragma unroll
    for (int jj = 0; jj < Nst; ++jj) acc += mrow[jj] * __shfl(s, jj, 16);
    s = acc;
  }
}

// Phase C: parallel replay of every chunk from its start state.
__global__ void scan_phaseC(const float* __restrict__ Ad, const float* __restrict__ Bu,
                            const float* __restrict__ cstart, float* __restrict__ st,
                            int Lr) {
  const int chunk = blockIdx.x * (blockDim.x >> 5) + (threadIdx.x >> 5);
  const int lane = threadIdx.x & 31;
  const int i = lane & 15;
  const int ncPerB = Lr / CLEN;
  const int b = chunk / ncPerB;
  const int c = chunk % ncPerB;
  const long t0 = (long)b * Lr + (long)c * CLEN;
  float s = cstart[(long)chunk * Nst + i];
  for (int t = 0; t < CLEN; ++t) {
    const long idx = t0 + t;
    float acc = Bu[idx * Nst + i];
    const float* arow = Ad + idx * (Nst * Nst) + (long)i * Nst;
#pragma unroll
    for (int jj = 0; jj < Nst; ++jj) acc += arow[jj] * __shfl(s, jj, 16);
    s = acc;
    if (lane < 16) st[idx * Nst + i] = s;
  }
}

// ---------------------------------------------------------------------------
// Fused C+D path with M=64 register blocking:
// y_pre[t,h] = sum_n (u@w_C+b_C)[t,h*N+n]*s[t,n] + sum_g (u@w_D+b_D)[t,h*H+g]*u[t,g]
// Each wave keeps MBLK=4 independent accumulators sharing one B load.
// ---------------------------------------------------------------------------
__global__ void cd_kernel(const float* __restrict__ u, const float* __restrict__ st,
                          const float* __restrict__ wC, const float* __restrict__ bC,
                          const float* __restrict__ wD, const float* __restrict__ bD,
                          float* __restrict__ ypre) {
  __shared__ float uLds[16 * MBLK * LDA];
  __shared__ float sLds[16 * MBLK * Nst];
  __shared__ float yAcc[16 * MBLK * 16];
  const int rowBase = blockIdx.x * (16 * MBLK);
  const int hBase = blockIdx.y * 16;
  loadTileRows(u, rowBase, 16 * MBLK, uLds);
  for (int idx = threadIdx.x; idx < 16 * MBLK * Nst; idx += blockDim.x) {
    const int r = idx >> 4, c = idx & 15;
    sLds[idx] = st[(long)(rowBase + r) * Nst + c];
  }
  for (int idx = threadIdx.x; idx < 16 * MBLK * 16; idx += blockDim.x) yAcc[idx] = 0.0f;
  __syncthreads();

  const int wave = threadIdx.x >> 5;
  const int lane = threadIdx.x & 31;
  const int m  = lane & 15;
  const int kh = (lane >> 4) << 1;
  const int j = lane & 15;
  const int roff = (lane >> 4) << 3;

  // D path: 128 column tiles (16 h x 8 g-chunks), 16 per wave, MBLK row tiles each.
  for (int ct = wave; ct < 128; ct += 8) {
    const int hl = ct >> 3;
    const int gc = ct & 7;
    const long colBase = (long)(hBase + hl) * Hdim + gc * 16;
    const float* wp = wD + colBase + j;
    v8f acc[MBLK];
#pragma unroll
    for (int s = 0; s < MBLK; ++s) acc[s] = (v8f){};
#pragma unroll
    for (int k = 0; k < Hdim; k += 4) {
      v2f b;
      b.x = wp[(long)(k + kh) * (Hdim * Hdim)];
      b.y = wp[(long)(k + kh + 1) * (Hdim * Hdim)];
#pragma unroll
      for (int s = 0; s < MBLK; ++s) {
        v2f a;
        a.x = uLds[(s * 16 + m) * LDA + k + kh];
        a.y = uLds[(s * 16 + m) * LDA + k + kh + 1];
        acc[s] = __builtin_amdgcn_wmma_f32_16x16x4_f32(false, a, false, b,
                                                       (short)0, acc[s], false, false);
      }
    }
    const float bb = bD[colBase + j];
#pragma unroll
    for (int s = 0; s < MBLK; ++s) {
#pragma unroll
      for (int r = 0; r < 8; ++r) {
        const int rr = s * 16 + r + roff;
        float val = (acc[s][r] + bb) * uLds[rr * LDA + gc * 16 + j];
        val = rowReduce16(val);
        if (j == 0) atomicAdd(&yAcc[rr * 16 + hl], val);
      }
    }
  }

  // C path: 16 column tiles (one per local h), 2 per wave, MBLK row tiles each.
  for (int ct = wave; ct < 16; ct += 8) {
    const long colBase = (long)(hBase + ct) * Nst;
    const float* wp = wC + colBase + j;
    v8f acc[MBLK];
#pragma unroll
    for (int s = 0; s < MBLK; ++s) acc[s] = (v8f){};
#pragma unroll
    for (int k = 0; k < Hdim; k += 4) {
      v2f b;
      b.x = wp[(long)(k + kh) * (Hdim * Nst)];
      b.y = wp[(long)(k + kh + 1) * (Hdim * Nst)];
#pragma unroll
      for (int s = 0; s < MBLK; ++s) {
        v2f a;
        a.x = uLds[(s * 16 + m) * LDA + k + kh];
        a.y = uLds[(s * 16 + m) * LDA + k + kh + 1];
        acc[s] = __builtin_amdgcn_wmma_f32_16x16x4_f32(false, a, false, b,
                                                       (short)0, acc[s], false, false);
      }
    }
    const float bb = bC[colBase + j];
#pragma unroll
    for (int s = 0; s < MBLK; ++s) {
#pragma unroll
      for (int r = 0; r < 8; ++r) {
        const int rr = s * 16 + r + roff;
        float val = (acc[s][r] + bb) * sLds[rr * Nst + j];
        val = rowReduce16(val);
        if (j == 0) atomicAdd(&yAcc[rr * 16 + ct], val);
      }
    }
  }

  __syncthreads();
  for (int idx = threadIdx.x; idx < 16 * MBLK * 16; idx += blockDim.x) {
    const int r = idx >> 4, c = idx & 15;
    ypre[(long)(rowBase + r) * Hdim + hBase + c] = yAcc[idx];
  }
}

// ---------------------------------------------------------------------------
extern "C" void kernel_launch(void* const* d_in, const int* in_sizes, int n_in,
                              void* d_out, int out_size, void* d_ws, size_t ws_size,
                              hipStream_t stream) {
  const float* x    = (const float*)d_in[0];
  const float* ln_g = (const float*)d_in[1];
  const float* ln_b = (const float*)d_in[2];
  const float* w_in = (const float*)d_in[3];
  const float* b_in = (const float*)d_in[4];
  const float* w_out= (const float*)d_in[5];
  const float* b_out= (const float*)d_in[6];
  const float* w_A  = (const float*)d_in[7];
  const float* b_A  = (const float*)d_in[8];
  const float* w_B  = (const float*)d_in[9];
  const float* b_B  = (const float*)d_in[10];
  const float* w_C  = (const float*)d_in[11];
  const float* b_C  = (const float*)d_in[12];
  const float* w_D  = (const float*)d_in[13];
  const float* b_D  = (const float*)d_in[14];
  const float* dt   = (const float*)d_in[15];

  const int rows = in_sizes[0] / Hdim;   // B*L = 8192
  const int Bb = 4;
  const int Lr = rows / Bb;
  const int rt = rows / 16;              // 512 row tiles
  const int ncPerB = Lr / CLEN;          // 64 chunks per batch
  const int nChunks = Bb * ncPerB;       // 256

  float* ws     = (float*)d_ws;
  float* xn     = ws;
  float* u      = xn + (size_t)rows * Hdim;
  float* Ad     = u  + (size_t)rows * Hdim;
  float* Bu     = Ad + (size_t)rows * (Nst * Nst);
  float* st     = Bu + (size_t)rows * Nst;
  float* ypre   = st + (size_t)rows * Nst;
  float* Mc     = ypre + (size_t)rows * Hdim;
  float* vc     = Mc + (size_t)nChunks * (Nst * Nst);
  float* cstart = vc + (size_t)nChunks * Nst;

  ln_kernel<<<rows, Hdim, 0, stream>>>(x, ln_g, ln_b, xn);
  gemm_k128<1><<<dim3(rt, 1), 256, 0, stream>>>(xn, w_in, b_in, nullptr, u, Hdim, dt);
  gemm_k128<2><<<dim3(rt, 2), 256, 0, stream>>>(u, w_A, b_A, nullptr, Ad, Nst * Nst, dt);
  bu_kernel<<<dim3(rt, Nst), 256, 0, stream>>>(u, w_B, b_B, Bu, dt);
  scan_phaseA<<<nChunks, 32, 0, stream>>>(Ad, Bu, Mc, vc, Lr);
  scan_phaseB<<<1, 32 * Bb, 0, stream>>>(Mc, vc, cstart, ncPerB);
  scan_phaseC<<<nChunks / 8, 256, 0, stream>>>(Ad, Bu, cstart, st, Lr);
  cd_kernel<<<dim3(rows / (16 * MBLK), Hdim / 16), 256, 0, stream>>>(
      u, st, w_C, b_C, w_D, b_D, ypre);
  gemm_k128<3><<<dim3(rt, 1), 256, 0, stream>>>(ypre, w_out, b_out, x, (float*)d_out, Hdim, dt);
}